// GCNLayer_65429531787486
// MI455X (gfx1250) — compile-verified
//
#include <hip/hip_runtime.h>
#include <hip/hip_bf16.h>

#define D 128
#define D4 32   // D / 4

typedef __attribute__((ext_vector_type(2))) float v2f;
typedef __attribute__((ext_vector_type(8))) float v8f;

// ---------------------------------------------------------------- zero fill
__global__ void gcn_zero_kernel(float* __restrict__ p, int n4) {
    int i = blockIdx.x * blockDim.x + threadIdx.x;
    if (i < n4) ((float4*)p)[i] = make_float4(0.f, 0.f, 0.f, 0.f);
}

// ---------------------------------------------------------------- degrees
__global__ void gcn_degree_kernel(const int* __restrict__ src, const int* __restrict__ dst,
                                  float* __restrict__ dout, float* __restrict__ din, int E) {
    int i = blockIdx.x * blockDim.x + threadIdx.x;
    if (i < E) {
        atomicAdd(&dout[src[i]], 1.0f);
        atomicAdd(&din[dst[i]], 1.0f);
    }
}

// ------------------------------------------- LayerNorm * out-degree norm
// one wave32 per row; 4 floats per lane (float4)
__global__ void gcn_ln_kernel(const float* __restrict__ feat, const float* __restrict__ deg_out,
                              const float* __restrict__ a2, const float* __restrict__ b2,
                              float* __restrict__ hn, int N) {
    int wave = threadIdx.x >> 5;
    int lane = threadIdx.x & 31;
    int row = blockIdx.x * (blockDim.x >> 5) + wave;
    if (row >= N) return;

    float4 x = ((const float4*)(feat + (size_t)row * D))[lane];
    float s  = x.x + x.y + x.z + x.w;
    float s2 = x.x * x.x + x.y * x.y + x.z * x.z + x.w * x.w;
    #pragma unroll
    for (int off = 16; off >= 1; off >>= 1) {
        s  += __shfl_xor(s,  off, 32);
        s2 += __shfl_xor(s2, off, 32);
    }
    float mean = s * (1.0f / 128.0f);
    float var  = (s2 - 128.0f * mean * mean) * (1.0f / 127.0f);
    var = fmaxf(var, 0.0f);
    float istd = 1.0f / (sqrtf(var) + 1e-6f);            // eps added to std (torch style)
    float no   = rsqrtf(fmaxf(deg_out[row], 1.0f));      // out-degree norm

    float4 g  = ((const float4*)a2)[lane];
    float4 bb = ((const float4*)b2)[lane];
    float4 r;
    r.x = (g.x * (x.x - mean) * istd + bb.x) * no;
    r.y = (g.y * (x.y - mean) * istd + bb.y) * no;
    r.z = (g.z * (x.z - mean) * istd + bb.z) * no;
    r.w = (g.w * (x.w - mean) * istd + bb.w) * no;
    ((float4*)(hn + (size_t)row * D))[lane] = r;
}

// ---------------------------------------------------- WMMA fp32 GEMM hn @ W
// one wave32 per 16x16 output tile; K=128 chained as 32x V_WMMA_F32_16X16X4_F32
__global__ void __launch_bounds__(128)
gcn_gemm_kernel(const float* __restrict__ A, const float* __restrict__ W,
                float* __restrict__ C, int N) {
    __shared__ float sW[D * D];   // 64 KB of the 320 KB/WGP LDS
    for (int i = threadIdx.x; i < D * D4; i += 128)
        ((float4*)sW)[i] = ((const float4*)W)[i];
    __syncthreads();

    int wave = threadIdx.x >> 5;
    int lane = threadIdx.x & 31;
    int rowTiles = (N + 15) >> 4;
    int tile = blockIdx.x * 4 + wave;          // wave-uniform guard (EXEC stays all-ones)
    if (tile >= rowTiles * 8) return;

    int rt = tile >> 3, ct = tile & 7;
    int m    = lane & 15;                      // M for A, N-col for B/C/D
    int half = lane >> 4;                      // K split (0: K+0/1, 1: K+2/3)
    int row  = rt * 16 + m;
    int arow = row < N ? row : N - 1;          // clamp loads, predicate stores
    const float* Ap = A + (size_t)arow * D + 2 * half;
    int col = ct * 16 + m;

    v8f c = {0.f, 0.f, 0.f, 0.f, 0.f, 0.f, 0.f, 0.f};
    #pragma unroll
    for (int kb = 0; kb < D; kb += 4) {
        v2f a = *(const v2f*)(Ap + kb);                       // 8B-aligned global_load_b64
        v2f b;
        b.x = sW[(kb + 2 * half + 0) * D + col];
        b.y = sW[(kb + 2 * half + 1) * D + col];
        c = __builtin_amdgcn_wmma_f32_16x16x4_f32(
                /*neg_a=*/false, a, /*neg_b=*/false, b,
                /*c_mod=*/(short)0, c, /*reuse_a=*/false, /*reuse_b=*/false);
    }
    #pragma unroll
    for (int r = 0; r < 8; r++) {
        int orow = rt * 16 + r + 8 * half;     // C/D layout: VGPR r -> M=r / M=r+8
        if (orow < N) C[(size_t)orow * D + col] = c[r];
    }
}

// ------------------------------------------------- edge scatter: agg[dst]+=hw[src]
// one wave32 per edge; float4 gather + 4 f32 atomic adds per lane
__global__ void gcn_scatter_kernel(const float* __restrict__ hw, const int* __restrict__ src,
                                   const int* __restrict__ dst, float* __restrict__ agg, int E) {
    int g    = blockIdx.x * blockDim.x + threadIdx.x;
    int e    = g >> 5;
    int lane = g & 31;
    if (e >= E) return;
    int s = src[e], d = dst[e];
    float4 v = ((const float4*)(hw + (size_t)s * D))[lane];
    float* p = agg + (size_t)d * D + lane * 4;
    atomicAdd(p + 0, v.x);
    atomicAdd(p + 1, v.y);
    atomicAdd(p + 2, v.z);
    atomicAdd(p + 3, v.w);
}

// --------------------------------------------- in-norm + bias + relu + residual
__global__ void gcn_final_kernel(const float* __restrict__ agg, const float* __restrict__ din,
                                 const float* __restrict__ b, const float* __restrict__ feat,
                                 float* __restrict__ out, int N) {
    int g = blockIdx.x * blockDim.x + threadIdx.x;      // one float4 per thread
    if (g >= N * D4) return;
    int row  = g >> 5;        // D4 == 32
    int lane = g & 31;
    float ni = rsqrtf(fmaxf(din[row], 1.0f));
    float4 v  = ((const float4*)agg)[g];
    float4 bb = ((const float4*)b)[lane];
    float4 f  = ((const float4*)feat)[g];
    float4 r;
    r.x = fmaxf(v.x * ni + bb.x, 0.f) + f.x;
    r.y = fmaxf(v.y * ni + bb.y, 0.f) + f.y;
    r.z = fmaxf(v.z * ni + bb.z, 0.f) + f.z;
    r.w = fmaxf(v.w * ni + bb.w, 0.f) + f.w;
    ((float4*)out)[g] = r;
}

extern "C" void kernel_launch(void* const* d_in, const int* in_sizes, int n_in,
                              void* d_out, int out_size, void* d_ws, size_t ws_size,
                              hipStream_t stream) {
    const float* feat = (const float*)d_in[0];
    const int*   src  = (const int*)  d_in[1];
    const int*   dst  = (const int*)  d_in[2];
    const float* W    = (const float*)d_in[3];
    const float* b    = (const float*)d_in[4];
    const float* a2   = (const float*)d_in[5];
    const float* b2   = (const float*)d_in[6];

    int N = in_sizes[0] / D;
    int E = in_sizes[1];

    // workspace layout (floats): deg_out[N] | deg_in[N] | hn/agg[N*D] | hw[N*D]
    float* ws      = (float*)d_ws;
    float* deg_out = ws;
    float* deg_in  = ws + N;
    float* hn      = ws + 2 * (size_t)N;       // reused as agg after GEMM consumes it
    float* hw      = hn + (size_t)N * D;
    float* agg     = hn;

    // 1) zero degree buffers (contiguous 2N floats)
    int dz4 = (2 * N) / 4;
    gcn_zero_kernel<<<(dz4 + 255) / 256, 256, 0, stream>>>(deg_out, dz4);

    // 2) degree histograms
    gcn_degree_kernel<<<(E + 255) / 256, 256, 0, stream>>>(src, dst, deg_out, deg_in, E);

    // 3) LayerNorm * out-degree norm  (8 rows per 256-thread block)
    gcn_ln_kernel<<<(N + 7) / 8, 256, 0, stream>>>(feat, deg_out, a2, b2, hn, N);

    // 4) hw = hn @ W via WMMA (4 tiles per 128-thread block)
    int tiles = ((N + 15) / 16) * 8;
    gcn_gemm_kernel<<<(tiles + 3) / 4, 128, 0, stream>>>(hn, W, hw, N);

    // 5) zero agg (reuse hn buffer)
    int a4 = N * D4;
    gcn_zero_kernel<<<(a4 + 255) / 256, 256, 0, stream>>>(agg, a4);

    // 6) edge scatter-add
    long sth = (long)E * 32;
    gcn_scatter_kernel<<<(int)((sth + 255) / 256), 256, 0, stream>>>(hw, src, dst, agg, E);

    // 7) final: relu(agg * in-norm + b) + feat
    int t = N * D4;
    gcn_final_kernel<<<(t + 255) / 256, 256, 0, stream>>>(agg, deg_in, b, feat, (float*)d_out, N);
}